// WeaveLayer_33105607918055
// MI455X (gfx1250) — compile-verified
//
#include <hip/hip_runtime.h>

typedef __attribute__((ext_vector_type(2))) float v2f;
typedef __attribute__((ext_vector_type(8))) float v8f;

#define NN 1024   // N nodes
#define F  64     // all feature dims are 64

// ---------------------------------------------------------------------------
// K1: XWi = X @ We[64:128] + be ; XWj = X @ We[128:192]
// 512 wave-tiles (2 matrices x 64 Mtiles x 4 Ntiles), 8 waves/block -> 64 blocks
// ---------------------------------------------------------------------------
__global__ __launch_bounds__(256)
void weave_proj_kernel(const float* __restrict__ X,
                       const float* __restrict__ We,   // 192 x 64
                       const float* __restrict__ be,   // 64
                       float* __restrict__ XWi,        // 1024 x 64 (bias folded in)
                       float* __restrict__ XWj)        // 1024 x 64
{
    const int tid  = threadIdx.x;
    const int wave = tid >> 5, lane = tid & 31;
    const int half = lane >> 4, l16 = lane & 15;
    const int t     = blockIdx.x * 8 + wave;     // 0..511
    const int which = t >> 8;                    // 0 -> Xi, 1 -> Xj
    const int rem   = t & 255;
    const int mt    = rem >> 2, nt = rem & 3;
    const int n0    = nt * 16;

    const float bias = (which == 0) ? be[n0 + l16] : 0.0f;
    v8f c = {bias, bias, bias, bias, bias, bias, bias, bias};

    // A: M = lane%16, K pair = {2*half, 2*half+1} of each 4-chunk
    const float* xrow = X + (size_t)(mt * 16 + l16) * F + 2 * half;
    // B: rows of We_Xi / We_Xj
    const float* wb = We + (size_t)(F * (1 + which) + 2 * half) * F + n0 + l16;
    #pragma unroll
    for (int k = 0; k < 16; ++k) {
        v2f a = *(const v2f*)(xrow + 4 * k);
        v2f b;
        b.x = wb[(size_t)(4 * k) * F];
        b.y = wb[(size_t)(4 * k + 1) * F];
        c = __builtin_amdgcn_wmma_f32_16x16x4_f32(false, a, false, b,
                                                  (short)0, c, false, false);
    }

    float* out = ((which == 0) ? XWi : XWj)
               + (size_t)(mt * 16 + 8 * half) * F + n0 + l16;
    #pragma unroll
    for (int v = 0; v < 8; ++v)
        out[(size_t)v * F] = c[v];
}

// ---------------------------------------------------------------------------
// K2: main edge GEMM + fused epilogue + row aggregation.
// One block per node i. 8 waves: (mtile = wave&1) x (ntile = wave>>1).
// Each pass covers 32 j's x all 64 outputs; 32 passes sweep j = 0..1023.
// All hot-loop addresses are base pointers bumped by a constant per pass, so
// every access lowers to a single global_load/store with an immediate offset.
// ---------------------------------------------------------------------------
__global__ __launch_bounds__(256)
void weave_edge_kernel(const float* __restrict__ E,     // 1024 x 1024 x 64
                       const int*   __restrict__ A,     // 1024 x 1024
                       const float* __restrict__ We,    // 192 x 64 (rows 0..63 = We_E)
                       const float* __restrict__ XWi,   // 1024 x 64 (bias folded)
                       const float* __restrict__ XWj,   // 1024 x 64
                       float* __restrict__ E_new,       // 1024 x 1024 x 64
                       float* __restrict__ E_agg)       // 1024 x 64
{
    __shared__ float accum[F];

    const int i    = blockIdx.x;
    const int tid  = threadIdx.x;
    const int wave = tid >> 5, lane = tid & 31;
    const int half = lane >> 4, l16 = lane & 15;
    const int mtile = wave & 1;                  // which 16-j sub-tile
    const int n0    = (wave >> 1) * 16;          // output sub-range

    if (tid < F) accum[tid] = 0.0f;
    __syncthreads();

    // Hoist We_E B-fragments (constant over the whole j sweep): 16 x v2f.
    v2f b[16];
    {
        const float* wb = We + (size_t)(2 * half) * F + n0 + l16;
        #pragma unroll
        for (int k = 0; k < 16; ++k) {
            b[k].x = wb[(size_t)(4 * k) * F];
            b[k].y = wb[(size_t)(4 * k + 1) * F];
        }
    }

    // XWi term: constant per (i, o) -> broadcast across all M rows of C.
    const float xi = XWi[(size_t)i * F + n0 + l16];

    // Hot-loop base pointers (bumped by 32 rows per pass).
    const float* erow = E + ((size_t)i * NN + (mtile * 16 + l16)) * F + 2 * half;
    const float* xwj  = XWj + (size_t)(mtile * 16 + 8 * half) * F + n0 + l16;
    const int*   arow = A + (size_t)i * NN + mtile * 16 + 8 * half;
    float*       eout = E_new + ((size_t)i * NN + (mtile * 16 + 8 * half)) * F
                              + n0 + l16;

    for (int pass = 0; pass < NN / 32; ++pass) {
        // Prefetch next pass's E rows (global_prefetch_b8) while we compute.
        if (pass != NN / 32 - 1) {
            __builtin_prefetch(erow + 32 * F, 0, 1);
            __builtin_prefetch(arow + 32, 0, 1);
        }

        // Issue epilogue operands early so they hide behind the WMMA chain.
        float xwjv[8], av[8];
        #pragma unroll
        for (int v = 0; v < 8; ++v) {
            xwjv[v] = xwj[(size_t)v * F];
            av[v]   = (float)arow[v];
        }

        v8f c = {xi, xi, xi, xi, xi, xi, xi, xi};
        #pragma unroll
        for (int k = 0; k < 16; ++k) {
            v2f a = *(const v2f*)(erow + 4 * k);
            c = __builtin_amdgcn_wmma_f32_16x16x4_f32(false, a, false, b[k],
                                                      (short)0, c, false, false);
        }

        // Fused epilogue: + XWj[j], relu, mask by A, store E_new, column sums.
        float colsum = 0.0f;
        #pragma unroll
        for (int v = 0; v < 8; ++v) {
            float e = fmaxf(c[v] + xwjv[v], 0.0f) * av[v];
            eout[(size_t)v * F] = e;
            colsum += e;
        }
        // combine the two lane halves (same N, disjoint M ranges)
        colsum += __shfl_xor(colsum, 16, 32);
        if (half == 0) atomicAdd(&accum[n0 + l16], colsum);

        erow += 32 * F;
        xwj  += 32 * F;
        arow += 32;
        eout += 32 * F;
    }

    __syncthreads();
    if (tid < F) E_agg[(size_t)i * F + tid] = accum[tid];
}

// ---------------------------------------------------------------------------
// K3: X_new = relu(X @ Wn[0:64] + E_agg @ Wn[64:128] + bn)
// 256 wave-tiles, 8 waves/block -> 32 blocks.
// ---------------------------------------------------------------------------
__global__ __launch_bounds__(256)
void weave_node_kernel(const float* __restrict__ X,
                       const float* __restrict__ Eagg,
                       const float* __restrict__ Wn,   // 128 x 64
                       const float* __restrict__ bn,   // 64
                       float* __restrict__ Xnew)       // 1024 x 64
{
    const int tid  = threadIdx.x;
    const int wave = tid >> 5, lane = tid & 31;
    const int half = lane >> 4, l16 = lane & 15;
    const int t  = blockIdx.x * 8 + wave;        // 0..255
    const int mt = t >> 2, nt = t & 3;
    const int n0 = nt * 16;

    const float bias = bn[n0 + l16];
    v8f c = {bias, bias, bias, bias, bias, bias, bias, bias};

    const float* xrow = X + (size_t)(mt * 16 + l16) * F + 2 * half;
    const float* wbx  = Wn + (size_t)(2 * half) * F + n0 + l16;
    #pragma unroll
    for (int k = 0; k < 16; ++k) {
        v2f a = *(const v2f*)(xrow + 4 * k);
        v2f b;
        b.x = wbx[(size_t)(4 * k) * F];
        b.y = wbx[(size_t)(4 * k + 1) * F];
        c = __builtin_amdgcn_wmma_f32_16x16x4_f32(false, a, false, b,
                                                  (short)0, c, false, false);
    }
    const float* arow = Eagg + (size_t)(mt * 16 + l16) * F + 2 * half;
    const float* wbe  = Wn + (size_t)(F + 2 * half) * F + n0 + l16;
    #pragma unroll
    for (int k = 0; k < 16; ++k) {
        v2f a = *(const v2f*)(arow + 4 * k);
        v2f b;
        b.x = wbe[(size_t)(4 * k) * F];
        b.y = wbe[(size_t)(4 * k + 1) * F];
        c = __builtin_amdgcn_wmma_f32_16x16x4_f32(false, a, false, b,
                                                  (short)0, c, false, false);
    }

    float* out = Xnew + (size_t)(mt * 16 + 8 * half) * F + n0 + l16;
    #pragma unroll
    for (int v = 0; v < 8; ++v)
        out[(size_t)v * F] = fmaxf(c[v], 0.0f);
}

// ---------------------------------------------------------------------------
extern "C" void kernel_launch(void* const* d_in, const int* in_sizes, int n_in,
                              void* d_out, int out_size, void* d_ws, size_t ws_size,
                              hipStream_t stream) {
    const float* X  = (const float*)d_in[0];   // 1024 x 64
    const float* E  = (const float*)d_in[1];   // 1024 x 1024 x 64
    const int*   A  = (const int*)  d_in[2];   // 1024 x 1024
    const float* We = (const float*)d_in[3];   // 192 x 64
    const float* be = (const float*)d_in[4];   // 64
    const float* Wn = (const float*)d_in[5];   // 128 x 64
    const float* bn = (const float*)d_in[6];   // 64

    // Output tuple (X_new, E_new) concatenated flat.
    float* Xnew = (float*)d_out;
    float* Enew = (float*)d_out + (size_t)NN * F;

    // Workspace: XWi (256KB) | XWj (256KB) | E_agg (256KB)
    float* XWi  = (float*)d_ws;
    float* XWj  = XWi + (size_t)NN * F;
    float* Eagg = XWj + (size_t)NN * F;

    weave_proj_kernel<<<64, 256, 0, stream>>>(X, We, be, XWi, XWj);
    weave_edge_kernel<<<NN, 256, 0, stream>>>(E, A, We, XWi, XWj, Enew, Eagg);
    weave_node_kernel<<<32, 256, 0, stream>>>(X, Eagg, Wn, bn, Xnew);
}